// SimVQ_Usage_41455024341704
// MI455X (gfx1250) — compile-verified
//
#include <hip/hip_runtime.h>

#define K_CB   8192
#define N_ROWS 8192
#define D_DIM  128
#define NTILES (K_CB / 16)   // 512 column tiles

typedef __attribute__((ext_vector_type(2))) float v2f;
typedef __attribute__((ext_vector_type(4))) float v4f;
typedef __attribute__((ext_vector_type(8))) float v8f;

// ---------------------------------------------------------------------------
// K0: zero counts + loss accumulator (d_ws is poisoned by harness)
// ---------------------------------------------------------------------------
__global__ void zero_ws_kernel(int* __restrict__ counts, float* __restrict__ loss) {
  int i = blockIdx.x * blockDim.x + threadIdx.x;
  if (i < K_CB) counts[i] = 0;
  if (i == 0) *loss = 0.0f;
}

// ---------------------------------------------------------------------------
// K1: qc = emb @ proj_w^T + proj_b ; qcn = row squared norms
// One block (128 threads) per codebook row.
// ---------------------------------------------------------------------------
__global__ __launch_bounds__(128)
void qc_kernel(const float* __restrict__ emb, const float* __restrict__ pw,
               const float* __restrict__ pb, float* __restrict__ qc,
               float* __restrict__ qcn) {
  __shared__ float se[D_DIM];
  __shared__ float red[D_DIM];
  const int k = blockIdx.x;
  const int d = threadIdx.x;
  se[d] = emb[k * D_DIM + d];
  __syncthreads();
  float acc = pb[d];
  const float* wrow = pw + d * D_DIM;
#pragma unroll 8
  for (int j = 0; j < D_DIM; ++j) acc = fmaf(se[j], wrow[j], acc);
  qc[k * D_DIM + d] = acc;
  red[d] = acc * acc;
  __syncthreads();
  for (int s = 64; s > 0; s >>= 1) {
    if (d < s) red[d] += red[d + s];
    __syncthreads();
  }
  if (d == 0) qcn[k] = red[0];
}

// ---------------------------------------------------------------------------
// K2: fused distance-GEMM + argmin via V_WMMA_F32_16X16X4_F32.
// Block = 128 threads = 4 waves; wave w owns z rows [ (blk*4+w)*16 .. +15 ).
// A (z tile) preloaded in registers for all 32 K-steps.
// B (qc tile, 16 cols x 128 depth) staged in LDS, row-padded to 132 dwords
// (132 mod 64 == 4 -> 16 lanes hit distinct bank groups, conflict-free b64).
// ---------------------------------------------------------------------------
__global__ __launch_bounds__(128)
void argmin_kernel(const float* __restrict__ z, const float* __restrict__ qc,
                   const float* __restrict__ qcn, int* __restrict__ idx_out) {
  __shared__ float tile[16 * 132];
  __shared__ float red_v[4 * 8 * 32];
  __shared__ int   red_i[4 * 8 * 32];

  const int t    = threadIdx.x;
  const int lane = t & 31;
  const int w    = t >> 5;
  const int row_tile = blockIdx.x * 4 + w;
  const int m    = lane & 15;        // row (A) / col (B) within tile
  const int koff = (lane >> 4) * 2;  // K sub-offset: lanes 0-15 -> K0/K1, 16-31 -> K2/K3

  // --- A operand preload: 32 k-steps x v2f = 64 VGPRs per lane ---
  v2f a[32];
  const float* zrow = z + (size_t)(row_tile * 16 + m) * D_DIM;
#pragma unroll
  for (int s = 0; s < 32; ++s)
    a[s] = *(const v2f*)(zrow + 4 * s + koff);

  float minv[8];
  int   mini[8];
#pragma unroll
  for (int r = 0; r < 8; ++r) { minv[r] = 3.4e38f; mini[r] = 0; }

  for (int ct = 0; ct < NTILES; ++ct) {
    // cooperative staged load: qc tile [16][128] -> LDS [16][132]
    const float* src = qc + (size_t)ct * 16 * D_DIM;
#pragma unroll
    for (int i = 0; i < 4; ++i) {
      int e  = (i * 128 + t) * 4;       // element index 0..2044 step 4
      int rr = e >> 7;                  // tile row
      int cc = e & 127;                 // tile col
      *(v4f*)&tile[rr * 132 + cc] = *(const v4f*)(src + e);
    }
    __syncthreads();

    // 32 chained f32 WMMAs: acc = z_tile (16x128) * qc_tile^T (128x16)
    v8f acc = {};
    const float* brow = &tile[m * 132 + koff];
#pragma unroll
    for (int s = 0; s < 32; ++s) {
      v2f b = *(const v2f*)(brow + 4 * s);
      acc = __builtin_amdgcn_wmma_f32_16x16x4_f32(
          false, a[s], false, b, (short)0, acc, false, false);
    }

    // distance (up to constant ||z||^2): qcn[col] - 2*dot ; running argmin
    const int   col = ct * 16 + m;
    const float qn  = qcn[col];
#pragma unroll
    for (int r = 0; r < 8; ++r) {
      float dist = fmaf(-2.0f, acc[r], qn);
      if (dist < minv[r]) { minv[r] = dist; mini[r] = col; }
    }
    __syncthreads();  // before next tile overwrites LDS
  }

  // cross-lane argmin: C layout -> VGPR r, lanes 0-15 = row r, lanes 16-31 = row r+8
#pragma unroll
  for (int r = 0; r < 8; ++r) {
    red_v[(w * 8 + r) * 32 + lane] = minv[r];
    red_i[(w * 8 + r) * 32 + lane] = mini[r];
  }
  __syncthreads();
  if (t < 64) {
    int w2 = t >> 4, mm = t & 15;
    int r = mm & 7, half = mm >> 3;
    int base = (w2 * 8 + r) * 32 + half * 16;
    float bv = red_v[base];
    int   bi = red_i[base];
#pragma unroll
    for (int j = 1; j < 16; ++j) {
      float v = red_v[base + j];
      if (v < bv) { bv = v; bi = red_i[base + j]; }
    }
    idx_out[(blockIdx.x * 4 + w2) * 16 + mm] = bi;
  }
}

// ---------------------------------------------------------------------------
// K3: z_q = qc[idx]; accumulate sum((z_q - z)^2); histogram counts
// One block (128 threads) per z row.
// ---------------------------------------------------------------------------
__global__ __launch_bounds__(128)
void gather_loss_kernel(const float* __restrict__ z, const float* __restrict__ qc,
                        const int* __restrict__ idx, float* __restrict__ zq_out,
                        int* __restrict__ counts, float* __restrict__ loss) {
  __shared__ float red[D_DIM];
  const int row = blockIdx.x;
  const int t   = threadIdx.x;
  const int k   = idx[row];
  float q  = qc[(size_t)k * D_DIM + t];
  float zv = z[(size_t)row * D_DIM + t];
  zq_out[(size_t)row * D_DIM + t] = q;   // straight-through forward == z_q_raw
  float d = q - zv;
  red[t] = d * d;
  __syncthreads();
  for (int s = 64; s > 0; s >>= 1) {
    if (t < s) red[t] += red[t + s];
    __syncthreads();
  }
  if (t == 0) {
    atomicAdd(loss, red[0]);
    atomicAdd(&counts[k], 1);
  }
}

// ---------------------------------------------------------------------------
// K4: scalars: commit_loss, perplexity, ema_entropy
// ---------------------------------------------------------------------------
__global__ __launch_bounds__(256)
void finalize_kernel(const int* __restrict__ counts, const float* __restrict__ ema,
                     const float* __restrict__ loss, float* __restrict__ out_scalars) {
  __shared__ float r0[256];
  __shared__ float r1[256];
  const int t = threadIdx.x;
  float eu = 0.0f, ee = 0.0f;
  for (int i = t; i < K_CB; i += 256) {
    float em = (float)counts[i] * (1.0f / (float)N_ROWS);
    float nu = ema[i] * 0.99f + em * 0.01f;
    eu -= nu * logf(nu + 1e-10f);
    ee -= em * logf(em + 1e-10f);
  }
  r0[t] = eu; r1[t] = ee;
  __syncthreads();
  for (int s = 128; s > 0; s >>= 1) {
    if (t < s) { r0[t] += r0[t + s]; r1[t] += r1[t + s]; }
    __syncthreads();
  }
  if (t == 0) {
    out_scalars[0] = 1.25f * (*loss) / (float)(N_ROWS * D_DIM);  // commit_loss
    out_scalars[1] = expf(r1[0]);                                // perplexity
    out_scalars[2] = r0[0];                                      // ema_entropy
  }
}

// ---------------------------------------------------------------------------
extern "C" void kernel_launch(void* const* d_in, const int* in_sizes, int n_in,
                              void* d_out, int out_size, void* d_ws, size_t ws_size,
                              hipStream_t stream) {
  (void)in_sizes; (void)n_in; (void)out_size; (void)ws_size;
  const float* z   = (const float*)d_in[0];   // [8,32,32,128]
  const float* emb = (const float*)d_in[1];   // [8192,128]
  const float* pw  = (const float*)d_in[2];   // [128,128]
  const float* pb  = (const float*)d_in[3];   // [128]
  const float* ema = (const float*)d_in[4];   // [8192]
  float* out = (float*)d_out;                 // z_q (1048576) + 3 scalars

  char* ws = (char*)d_ws;
  float* qc   = (float*)(ws);                                   // 4 MB
  float* qcn  = (float*)(ws + 4u * 1024 * 1024);                // 32 KB
  int*   idx  = (int*)  (ws + 4u * 1024 * 1024 + 32 * 1024);    // 32 KB
  int*   cnt  = (int*)  (ws + 4u * 1024 * 1024 + 64 * 1024);    // 32 KB
  float* loss = (float*)(ws + 4u * 1024 * 1024 + 96 * 1024);    // 4 B

  zero_ws_kernel<<<(K_CB + 255) / 256, 256, 0, stream>>>(cnt, loss);
  qc_kernel<<<K_CB, 128, 0, stream>>>(emb, pw, pb, qc, qcn);
  argmin_kernel<<<N_ROWS / 64, 128, 0, stream>>>(z, qc, qcn, idx);
  gather_loss_kernel<<<N_ROWS, 128, 0, stream>>>(z, qc, idx, out, cnt, loss);
  finalize_kernel<<<1, 256, 0, stream>>>(cnt, ema, loss, out + (size_t)N_ROWS * D_DIM);
}